// InitialModel_69337952026771
// MI455X (gfx1250) — compile-verified
//
#include <hip/hip_runtime.h>
#include <stdint.h>

// ---------------------------------------------------------------------------
// x[b,t,:] = x0[b,:] + SIGMA * cumsum(eps, t)    (x[b,0,:] = x0)
// Batched prefix sum done as chained V_WMMA_F32_16X16X4_F32 against an
// upper-triangular ones matrix, fed by async global->LDS copies (gfx1250).
// ---------------------------------------------------------------------------

#define NB      8192
#define TT      4096
#define TM1     4095
#define NCHUNK  256          // 256 chunks of 16 output timesteps (t = 16c+1 .. 16c+16)

typedef __attribute__((ext_vector_type(2))) float v2f;
typedef __attribute__((ext_vector_type(8))) float v8f;
typedef __attribute__((ext_vector_type(2))) int   v2i;

// typed, address-space-qualified pointers
typedef __attribute__((address_space(1))) v2i*   g_v2i_p;   // async builtin global arg
typedef __attribute__((address_space(3))) v2i*   l_v2i_p;   // async builtin LDS arg
typedef __attribute__((address_space(3))) float* l_f32_p;   // ds_read path for A

__device__ __forceinline__ void async_ld_b64(unsigned lds_byte_off, unsigned long long gaddr) {
#if __has_builtin(__builtin_amdgcn_global_load_async_to_lds_b64)
  __builtin_amdgcn_global_load_async_to_lds_b64(
      (g_v2i_p)(uintptr_t)gaddr, (l_v2i_p)(uintptr_t)(size_t)lds_byte_off, 0, 0);
#else
  asm volatile("global_load_async_to_lds_b64 %0, %1, off"
               :: "v"(lds_byte_off), "v"(gaddr) : "memory");
#endif
}

__device__ __forceinline__ void wait_async_le4() {
#if __has_builtin(__builtin_amdgcn_s_wait_asynccnt)
  __builtin_amdgcn_s_wait_asynccnt(4);
#else
  asm volatile("s_wait_asynccnt 4" ::: "memory");
#endif
  asm volatile("" ::: "memory");   // keep LDS reads below the wait
}

__device__ __forceinline__ void wait_async_le0() {
#if __has_builtin(__builtin_amdgcn_s_wait_asynccnt)
  __builtin_amdgcn_s_wait_asynccnt(0);
#else
  asm volatile("s_wait_asynccnt 0" ::: "memory");
#endif
  asm volatile("" ::: "memory");
}

__device__ __forceinline__ v8f wmma_f32_16x16x4(v2f a, v2f b, v8f c) {
  // 8-arg pattern: (neg_a, A, neg_b, B, c_mod, C, reuse_a, reuse_b)
  return __builtin_amdgcn_wmma_f32_16x16x4_f32(false, a, false, b, (short)0, c, false, false);
}

__global__ __launch_bounds__(256) void scan_wmma_kernel(
    const float* __restrict__ eps0,   // [NB, 2]
    const float* __restrict__ eps,    // [NB, TM1, 2]
    float* __restrict__ out)          // [NB, TT, 2]
{
  // 8 waves/block, each wave owns 8 rows (= 16 sequences: row x channel).
  __shared__ float lds[8 * 2 * 256];              // per wave: 2 buffers x 1KB tile

  // Launder a 32-bit LDS *byte offset* (0 = base of our one shared array)
  // through inline asm, keeping the array itself alive/escaped via the input
  // operand. Reads via an AS(3) pointer built from this opaque offset lower
  // to ds_load, conservatively alias the async engine's LDS writes, and the
  // "memory" barriers after each s_wait_asynccnt pin them per-chunk.
  unsigned ldsBase = 0;
  asm volatile("" : "+v"(ldsBase) : "v"(lds));
  const l_f32_p L = (l_f32_p)(uintptr_t)(size_t)ldsBase;   // AS3 float* (ds_read path)

  const int lane   = threadIdx.x & 31;
  const int waveIB = threadIdx.x >> 5;
  const long long b0 = ((long long)blockIdx.x * 8 + waveIB) * 8;   // first row of this wave

  const float MEANX = 100.0f, MEANY = 75.0f;
  const float STDX  = 57.735026918962576f;        // 200/sqrt(12)
  const float STDY  = 43.301270189221932f;        // 150/sqrt(12)
  const float SIG   = 0.1f, INVSIG = 10.0f;

  // ---- carries: carry[i] corresponds to matrix row M = i + (lane>=16 ? 8 : 0)
  const int mhi = (lane >> 4) << 3;
  float carry[8];
#pragma unroll
  for (int i = 0; i < 8; ++i) {
    const int  M  = i + mhi;
    const long long b = b0 + (M >> 1);
    const int  ch = M & 1;
    const float e0 = eps0[b * 2 + ch];
    const float x0 = ch ? (MEANY + STDY * e0) : (MEANX + STDX * e0);
    carry[i] = x0 * INVSIG;                       // out = SIG * D  =>  seed carry with x0/SIG
  }

  // ---- x[:,0,:] = x0  (lanes 0..15 cover the 16 sequences)
  if (lane < 16) {
    const long long b = b0 + (lane >> 1);
    const int  ch = lane & 1;
    const float e0 = eps0[b * 2 + ch];
    out[(b * TT) * 2 + ch] = ch ? (MEANY + STDY * e0) : (MEANX + STDX * e0);
  }

  // ---- constant B operand: upper-triangular ones U[k,n] = (k<=n), K-sliced by 4.
  // B 4x16 layout (mirrors A): VGPR0: K = 4j+(lane<16?0:2), VGPR1: +1, N = lane&15.
  const int n   = lane & 15;
  const int kbh = (lane & 16) ? 2 : 0;
  v2f bmat[4];
#pragma unroll
  for (int j = 0; j < 4; ++j) {
    const int k0 = 4 * j + kbh;
    bmat[j][0] = (k0     <= n) ? 1.0f : 0.0f;
    bmat[j][1] = (k0 + 1 <= n) ? 1.0f : 0.0f;
  }

  // ---- async copy setup: per chunk a row contributes 128B contiguous (16 t x 2 ch x 4B).
  const unsigned ldsWaveBase = ldsBase + (unsigned)(waveIB * 2048);   // bytes
  const int rr = lane >> 2;                                           // row-in-wave 0..7
  const int pp = lane & 3;                                            // 8B-piece lane id
  const unsigned long long rowBase =
      (unsigned long long)(uintptr_t)eps + (unsigned long long)(b0 + rr) * (TM1 * 8ull);

  auto issueChunk = [&](int c) {
    const unsigned lbase = ldsWaveBase + ((unsigned)(c & 1)) * 1024u + (unsigned)rr * 128u;
    const unsigned long long gbase = rowBase + (unsigned long long)c * 128ull;
#pragma unroll
    for (int g = 0; g < 4; ++g) {                 // 4 async b64 instrs / chunk (32 lanes x 8B)
      const int piece = g * 4 + pp;               // 16 pieces of 8B per row
      if (!(c == NCHUNK - 1 && piece == 15))      // t=4095 doesn't exist; mask that 8B
        async_ld_b64(lbase + (unsigned)piece * 8u,
                     gbase + (unsigned long long)piece * 8ull);
    }
  };

  issueChunk(0);

  const int m    = lane & 15;                     // A-matrix row for this lane
  const int rloc = m >> 1;
  const int chA  = m & 1;
  const unsigned tileBaseA = (unsigned)(waveIB * 512) + (unsigned)(rloc * 32) + (unsigned)chA;

  for (int c = 0; c < NCHUNK; ++c) {
    const bool last = (c == NCHUNK - 1);
    if (!last) { issueChunk(c + 1); wait_async_le4(); }
    else       { wait_async_le0(); }

    // ---- build A (16x16 data, K-sliced by 4) from the LDS tile for chunk c
    const unsigned base = tileBaseA + ((unsigned)(c & 1)) * 256u;   // float index
    v2f a[4];
#pragma unroll
    for (int j = 0; j < 4; ++j) {
      const int k0 = 4 * j + kbh;                 // global k in 0..15 (t_e = 16c + k)
      float a0 = L[base + (unsigned)(k0)     * 2u];
      float a1 = L[base + (unsigned)(k0 + 1) * 2u];
      if (last && k0 == 14) a1 = 0.0f;            // eps[4095] doesn't exist
      a[j][0] = a0;
      a[j][1] = a1;
    }

    // ---- D = U-scan of chunk + carry  (4 chained fp32 WMMAs, K=16 total)
    v8f acc;
#pragma unroll
    for (int i = 0; i < 8; ++i) acc[i] = carry[i];
    acc = wmma_f32_16x16x4(a[0], bmat[0], acc);
    acc = wmma_f32_16x16x4(a[1], bmat[1], acc);
    acc = wmma_f32_16x16x4(a[2], bmat[2], acc);
    acc = wmma_f32_16x16x4(a[3], bmat[3], acc);

    // ---- carry update: new carry[M] = D[M,15]  (lane 15 for M<8, lane 31 for M>=8)
    const int srcLane = (lane & 16) | 15;
#pragma unroll
    for (int i = 0; i < 8; ++i) carry[i] = __shfl(acc[i], srcLane, 32);

    // ---- store x[b, 16c+n+1, ch] = SIG * D[M,n]
    if (!(last && n == 15)) {
      const long long tOut = 16LL * c + n + 1;
#pragma unroll
      for (int i = 0; i < 8; ++i) {
        const int  M  = i + mhi;
        const long long b = b0 + (M >> 1);
        const int  ch = M & 1;
        out[(b * TT + tOut) * 2 + ch] = SIG * acc[i];
      }
    }
  }
}

extern "C" void kernel_launch(void* const* d_in, const int* in_sizes, int n_in,
                              void* d_out, int out_size, void* d_ws, size_t ws_size,
                              hipStream_t stream) {
  const float* eps0 = (const float*)d_in[0];   // [8192, 2]
  const float* eps  = (const float*)d_in[1];   // [8192, 4095, 2]
  // d_in[2] = mini_batch_length: does not affect the output (log-prob mask only)
  float* out = (float*)d_out;                  // [8192, 4096, 2]

  // 8 rows/wave * 8 waves/block = 64 rows/block -> 128 blocks of 256 threads
  scan_wmma_kernel<<<dim3(NB / 64), dim3(256), 0, stream>>>(eps0, eps, out);
}